// Capsule_54657753809237
// MI455X (gfx1250) — compile-verified
//
#include <hip/hip_runtime.h>
#include <hip/hip_bf16.h>

typedef __attribute__((ext_vector_type(2))) float v2f;
typedef __attribute__((ext_vector_type(8))) float v8f;

#define N_IN    1152
#define IN_DIM  8
#define N_OUT   10
#define OUT_DIM 16
#define MDIM    160   /* N_OUT * OUT_DIM */
#define BATCH   256

// ---------------------------------------------------------------------------
// Kernel 1: uhat[b][n][m] = sum_i u[b][n][i] * W[n][i][m]   (fp32 WMMA)
// grid = (N_IN, BATCH/16), block = 320 threads (10 waves; wave w owns m-tile w)
// Each wave computes one 16(batch) x 16(m) tile via two V_WMMA_F32_16X16X4_F32
// (K=8 split into 2 x K=4). Result stored bf16 so the 94 MB uhat tensor stays
// resident in the 192 MB L2 for all 5 routing passes.
// ---------------------------------------------------------------------------
__global__ __launch_bounds__(320)
void capsule_uhat_wmma(const float* __restrict__ u,
                       const float* __restrict__ W,
                       __hip_bfloat16* __restrict__ uhat) {
  const int n    = blockIdx.x;        // input capsule
  const int mt   = blockIdx.y;        // batch tile of 16
  const int wave = threadIdx.x >> 5;  // 0..9 -> m-tile
  const int lane = threadIdx.x & 31;
  const int half = lane >> 4;         // 0: lanes 0-15, 1: lanes 16-31
  const int l16  = lane & 15;

  // A operand, 16x4 f32 (ISA 7.12.2): lanes0-15 hold K={0,1}, lanes16-31 K={2,3}
  const int    brow = mt * 16 + l16;
  const float* urow = u + ((size_t)brow * N_IN + n) * IN_DIM;
  const int    kb   = 2 * half;
  v2f a0, a1;
  {
    float2 t0 = *(const float2*)(urow + kb);       // K = kb, kb+1
    float2 t1 = *(const float2*)(urow + kb + 4);   // K = kb+4, kb+5
    a0[0] = t0.x; a0[1] = t0.y;
    a1[0] = t1.x; a1[1] = t1.y;
  }

  // B operand, 4x16 f32: lanes0-15 rows {0,1}, lanes16-31 rows {2,3}; col = l16
  const int    m     = wave * 16 + l16;
  const float* wbase = W + (size_t)n * IN_DIM * MDIM + m;
  v2f b0, b1;
  b0[0] = wbase[(kb + 0) * MDIM];
  b0[1] = wbase[(kb + 1) * MDIM];
  b1[0] = wbase[(kb + 4) * MDIM];
  b1[1] = wbase[(kb + 5) * MDIM];

  v8f c = {};
  // (neg_a, A, neg_b, B, c_mod, C, reuse_a, reuse_b)
  c = __builtin_amdgcn_wmma_f32_16x16x4_f32(false, a0, false, b0, (short)0, c,
                                            false, false);
  c = __builtin_amdgcn_wmma_f32_16x16x4_f32(false, a1, false, b1, (short)0, c,
                                            false, false);

  // C/D layout: VGPR j -> row M=j (lanes0-15) / M=j+8 (lanes16-31); col = l16
#pragma unroll
  for (int j = 0; j < 8; ++j) {
    const int row = mt * 16 + j + 8 * half;
    uhat[((size_t)row * N_IN + n) * MDIM + m] = __float2bfloat16(c[j]);
  }
}

// ---------------------------------------------------------------------------
// Kernel 2: fused dynamic routing (3 iterations), one block per batch element.
// 640 threads = 4 n-groups x 160 (o,d) slots. Logits + softmax stats + partial
// sums in LDS (~58 KB static). uhat is read bf16 (L2 resident), accumulation
// in fp32.
// ---------------------------------------------------------------------------
__global__ __launch_bounds__(640)
void capsule_route(const __hip_bfloat16* __restrict__ uhat,
                   float* __restrict__ vout) {
  __shared__ float bl[N_IN * N_OUT];   // routing logits b   (46080 B)
  __shared__ float mx[N_IN];           // per-n softmax max
  __shared__ float rden[N_IN];         // per-n 1/sum(exp)
  __shared__ float spart[4 * MDIM];    // per-group s partials
  __shared__ float sfin[MDIM];         // reduced s
  __shared__ float vbuf[MDIM];         // squashed v
  __shared__ float scl[N_OUT];         // squash scale per o

  const int b   = blockIdx.x;
  const int tid = threadIdx.x;
  const __hip_bfloat16* ub = uhat + (size_t)b * N_IN * MDIM;

  for (int i = tid; i < N_IN * N_OUT; i += 640) bl[i] = 0.0f;
  __syncthreads();

  const int g = tid / MDIM;   // n-group 0..3
  const int t = tid % MDIM;   // flattened (o,d)
  const int o = t / OUT_DIM;

  for (int r = 0; r < 3; ++r) {
    // ---- per-n softmax stats (max + 1/sum of exp over the 10 logits) ----
    for (int n = tid; n < N_IN; n += 640) {
      float m0 = bl[n * N_OUT];
#pragma unroll
      for (int k = 1; k < N_OUT; ++k) m0 = fmaxf(m0, bl[n * N_OUT + k]);
      float s = 0.0f;
#pragma unroll
      for (int k = 0; k < N_OUT; ++k) s += __expf(bl[n * N_OUT + k] - m0);
      mx[n]   = m0;
      rden[n] = 1.0f / s;
    }
    __syncthreads();

    // ---- s[o,d] = sum_n c[n,o] * uhat[n,o,d] ; group g handles n % 4 == g ----
    float acc = 0.0f;
    for (int n = g; n < N_IN; n += 4) {
      const float cno = __expf(bl[n * N_OUT + o] - mx[n]) * rden[n];
      acc += cno * __bfloat162float(ub[(size_t)n * MDIM + t]);
    }
    spart[g * MDIM + t] = acc;
    __syncthreads();

    if (tid < MDIM)
      sfin[tid] = spart[tid] + spart[MDIM + tid] + spart[2 * MDIM + tid] +
                  spart[3 * MDIM + tid];
    __syncthreads();

    // ---- squash: scale_o = |s|^2 / ((1+|s|^2) * |s|) ----
    if (tid < N_OUT) {
      float n2 = 0.0f;
#pragma unroll
      for (int d = 0; d < OUT_DIM; ++d) {
        const float x = sfin[tid * OUT_DIM + d];
        n2 += x * x;
      }
      scl[tid] = n2 / ((1.0f + n2) * sqrtf(n2));
    }
    __syncthreads();

    if (tid < MDIM) {
      const float vv = scl[tid / OUT_DIM] * sfin[tid];
      vbuf[tid] = vv;
      if (r == 2) vout[(size_t)b * MDIM + tid] = vv;  // final output
    }
    __syncthreads();

    // ---- logit update: b[n,o] += sum_d uhat[n,o,d] * v[o,d] ----
    if (r < 2) {
      for (int p = tid; p < N_IN * N_OUT; p += 640) {
        const int nn = p / N_OUT;
        const int oo = p % N_OUT;
        const __hip_bfloat16* uo = ub + (size_t)nn * MDIM + oo * OUT_DIM;
        float a = 0.0f;
#pragma unroll
        for (int d = 0; d < OUT_DIM; ++d)
          a += __bfloat162float(uo[d]) * vbuf[oo * OUT_DIM + d];
        bl[p] += a;
      }
    }
    __syncthreads();
  }
}

// ---------------------------------------------------------------------------
extern "C" void kernel_launch(void* const* d_in, const int* in_sizes, int n_in,
                              void* d_out, int out_size, void* d_ws,
                              size_t ws_size, hipStream_t stream) {
  const float* u = (const float*)d_in[0];  // [256, 1152, 8]
  const float* W = (const float*)d_in[1];  // [1152, 8, 160]
  __hip_bfloat16* uhat = (__hip_bfloat16*)d_ws;  // [256, 1152, 160] bf16, 94 MB

  dim3 g1(N_IN, BATCH / 16);
  capsule_uhat_wmma<<<g1, 320, 0, stream>>>(u, W, uhat);
  capsule_route<<<BATCH, 640, 0, stream>>>(uhat, (float*)d_out);
}